// SSELayer_78795470013037
// MI455X (gfx1250) — compile-verified
//
#include <hip/hip_runtime.h>

// SSE / squeeze-excite fused layer for MI455X (gfx1250), fp32 via
// V_WMMA_F32_16X16X4_F32. One wave32 handles a 16-pixel group.
//
// x  : [B=32, C=256, H=64, W=64] f32
// W1 : [S=64, C=256], b1:[64]
// W2 : [C=256, S=64], b2:[256]
// out = x * hardsigmoid(W2 @ relu(W1 @ x + b1) + b2)   (per pixel)

typedef __attribute__((ext_vector_type(2))) float v2f;
typedef __attribute__((ext_vector_type(8))) float v8f;

#define C_IN    256
#define S_MID   64
#define HW      4096          // H*W, pixels per image
#define NIMG    32
#define WAVES_PER_BLOCK 8
#define Y_PITCH 20            // LDS row pitch (floats) for y scratch; bank-conflict-free

static __device__ __forceinline__ v8f wmma_f32(v2f a, v2f b, v8f c) {
  // D = A(16x4 f32) * B(4x16 f32) + C(16x16 f32)
  return __builtin_amdgcn_wmma_f32_16x16x4_f32(
      /*neg_a=*/false, a, /*neg_b=*/false, b,
      /*c_mod=*/(short)0, c, /*reuse_a=*/false, /*reuse_b=*/false);
}

__global__ __launch_bounds__(256)
void sse_fused_kernel(const float* __restrict__ x,
                      const float* __restrict__ W1,
                      const float* __restrict__ b1,
                      const float* __restrict__ W2,
                      const float* __restrict__ b2,
                      float* __restrict__ out) {
  // Per-wave y scratch: 64 rows (S) x 16 pixels, padded pitch. 8*64*20*4B = 40KB.
  __shared__ float ybuf[WAVES_PER_BLOCK * S_MID * Y_PITCH];

  const int lane = threadIdx.x & 31;
  const int wave = threadIdx.x >> 5;
  const int n    = lane & 15;   // pixel column within the 16-pixel group
  const int hi   = lane >> 4;   // half-wave (selects K pair / row-half per WMMA layout)

  const int g   = blockIdx.x * WAVES_PER_BLOCK + wave;  // group id, 0..8191
  const int img = g >> 8;                               // 256 groups per image
  const int p0  = (g & 255) << 4;                       // pixel base within image

  const float* xcol = x   + (size_t)img * (C_IN * HW) + p0 + n;  // + c*HW
  float*       ocol = out + (size_t)img * (C_IN * HW) + p0 + n;

  // ---------------- Stage 1: y[64x16] = relu(W1 @ x_col + b1) ----------------
  // A = W1 tile (16x4), B = x (4 channels x 16 pixels). 4 S-tiles, K=256 in steps of 4.
  v8f acc[4] = {};
  for (int kc = 0; kc < C_IN / 4; ++kc) {
    const int k0 = kc * 4 + hi * 2;             // this half-wave's K pair
    v2f bfrag;                                  // B: b[k,n] = x[k0+j, pixel n]
    bfrag.x = xcol[(size_t)(k0 + 0) * HW];
    bfrag.y = xcol[(size_t)(k0 + 1) * HW];
#pragma unroll
    for (int t = 0; t < 4; ++t) {
      // A: a[m,k] = W1[t*16 + m, k0 + j]; rows are 1KB, k0 even -> 8B aligned
      v2f afrag = *reinterpret_cast<const v2f*>(W1 + (t * 16 + n) * C_IN + k0);
      acc[t] = wmma_f32(afrag, bfrag, acc[t]);
    }
  }

  // Bias + ReLU, relayout through LDS (C-layout -> B-layout needs cross-lane move).
  float* ys = &ybuf[wave * S_MID * Y_PITCH];
#pragma unroll
  for (int t = 0; t < 4; ++t) {
#pragma unroll
    for (int j = 0; j < 8; ++j) {
      const int s = t * 16 + hi * 8 + j;        // D layout: lane holds rows j + 8*hi
      float v = acc[t][j] + b1[s];
      ys[s * Y_PITCH + n] = fmaxf(v, 0.0f);
    }
  }
  __syncthreads();  // cross-lane LDS dependency within each wave

  // ------- Stage 2: s = hardsigmoid(W2 @ y + b2); out = x * s ---------------
  // A = W2 tile (16x4 of S), B = y from LDS. 16 C-tiles, K=64 in steps of 4.
  for (int ct = 0; ct < 16; ++ct) {
    v8f d = {};
#pragma unroll 4
    for (int ks = 0; ks < S_MID / 4; ++ks) {
      const int k0 = ks * 4 + hi * 2;
      // W2 rows are 256B, k0 even -> 8B aligned
      v2f afrag = *reinterpret_cast<const v2f*>(W2 + (ct * 16 + n) * S_MID + k0);
      v2f bfrag;
      bfrag.x = ys[(k0 + 0) * Y_PITCH + n];
      bfrag.y = ys[(k0 + 1) * Y_PITCH + n];
      d = wmma_f32(afrag, bfrag, d);
    }
#pragma unroll
    for (int j = 0; j < 8; ++j) {
      const int c = ct * 16 + hi * 8 + j;
      float sv = d[j] + b2[c];
      // hardsigmoid: clip(x/6 + 0.5, 0, 1)
      sv = fminf(fmaxf(sv * 0.16666667f + 0.5f, 0.0f), 1.0f);
      const size_t off = (size_t)c * HW;
      ocol[off] = xcol[off] * sv;   // x re-read hits L2 (block working set ~128KB)
    }
  }
}

extern "C" void kernel_launch(void* const* d_in, const int* in_sizes, int n_in,
                              void* d_out, int out_size, void* d_ws, size_t ws_size,
                              hipStream_t stream) {
  const float* x  = (const float*)d_in[0];
  const float* W1 = (const float*)d_in[1];
  const float* b1 = (const float*)d_in[2];
  const float* W2 = (const float*)d_in[3];
  const float* b2 = (const float*)d_in[4];
  float* out = (float*)d_out;

  const int total_groups = NIMG * (HW / 16);              // 8192 waves of work
  const int blocks = total_groups / WAVES_PER_BLOCK;      // 1024 blocks x 256 threads
  sse_fused_kernel<<<blocks, 256, 0, stream>>>(x, W1, b1, W2, b2, out);
}